// GraphSAGEConvLayer_5592047419417
// MI455X (gfx1250) — compile-verified
//
#include <hip/hip_runtime.h>
#include <hip/hip_bf16.h>

typedef __attribute__((ext_vector_type(2))) float v2f;
typedef __attribute__((ext_vector_type(8))) float v8f;

#define IN_CH   64
#define KDIM    128
#define OUT_CH  64

// ---------------------------------------------------------------------------
// Kernel 1: zero the workspace (aggr + deg). Must run every launch since the
// harness does not re-poison between replays and we accumulate atomically.
// ---------------------------------------------------------------------------
__global__ void sage_zero_ws(float* __restrict__ p, long long total) {
    long long i = (long long)blockIdx.x * blockDim.x + threadIdx.x;
    if (i < total) p[i] = 0.0f;
}

// ---------------------------------------------------------------------------
// Kernel 2: edge scatter-add. 16 threads per edge; each thread handles 4
// contiguous channels with a float4 gather + 4 float atomics.
// ---------------------------------------------------------------------------
__global__ void sage_scatter(const float* __restrict__ x,
                             const int* __restrict__ edges,
                             float* __restrict__ aggr,
                             float* __restrict__ deg,
                             int num_edges) {
    long long gid = (long long)blockIdx.x * blockDim.x + threadIdx.x;
    int e    = (int)(gid >> 4);
    int part = (int)(gid & 15);
    if (e >= num_edges) return;

    int src = edges[e];              // edge_index[0][e]
    int dst = edges[num_edges + e];  // edge_index[1][e]

    int c = part << 2;
    const float4 v = *(const float4*)(x + (size_t)src * IN_CH + c);
    float* d = aggr + (size_t)dst * IN_CH + c;
    atomicAdd(d + 0, v.x);
    atomicAdd(d + 1, v.y);
    atomicAdd(d + 2, v.z);
    atomicAdd(d + 3, v.w);
    if (part == 0) atomicAdd(&deg[dst], 1.0f);
}

// ---------------------------------------------------------------------------
// Kernel 3: mean normalization, aggr[i][c] /= max(deg[i], 1)
// ---------------------------------------------------------------------------
__global__ void sage_normalize(float* __restrict__ aggr,
                               const float* __restrict__ deg,
                               long long total) {
    long long i = (long long)blockIdx.x * blockDim.x + threadIdx.x;
    if (i >= total) return;
    float dg = deg[i >> 6];  // i / IN_CH
    aggr[i] = aggr[i] / fmaxf(dg, 1.0f);
}

// ---------------------------------------------------------------------------
// Kernel 4: WMMA GEMM. out[m][n] = sum_k conc[m][k] * W[n][k] + b[n]
// conc = [x | aggr], K = 128, N = 64. One wave computes a 16x64 tile using
// V_WMMA_F32_16X16X4_F32 (exact fp32), 32 K-steps x 4 N-tiles = 128 WMMAs.
// ---------------------------------------------------------------------------
static __device__ __forceinline__ v2f ld2(const float* p) {
    return *(const v2f*)p;
}

static __device__ __forceinline__ v8f wmma_f32k4(v2f a, v2f b, v8f c) {
    // (neg_a, A, neg_b, B, c_mod, C, reuse_a, reuse_b)
    return __builtin_amdgcn_wmma_f32_16x16x4_f32(false, a, false, b,
                                                 (short)0, c, false, false);
}

__global__ void sage_gemm_wmma(const float* __restrict__ x,
                               const float* __restrict__ aggr,
                               const float* __restrict__ W,
                               const float* __restrict__ bias,
                               float* __restrict__ out,
                               int num_tiles) {
    const int wave = threadIdx.x >> 5;
    const int lane = threadIdx.x & 31;
    const int tile = blockIdx.x * (blockDim.x >> 5) + wave;
    if (tile >= num_tiles) return;  // wave-uniform exit, EXEC stays all-ones

    const int m  = lane & 15;        // row within tile (A) / col N (B, C/D)
    const int ko = (lane >> 4) << 1; // K sub-offset: 0 for lanes 0-15, 2 for 16-31
    const int node = tile * 16 + m;

    const float* xrow = x    + (size_t)node * IN_CH;
    const float* arow = aggr + (size_t)node * IN_CH;
    // B fragment source rows: W[n][k], n = t*16 + (lane&15)
    const float* w0 = W + (size_t)(0 * 16 + m) * KDIM;
    const float* w1 = W + (size_t)(1 * 16 + m) * KDIM;
    const float* w2 = W + (size_t)(2 * 16 + m) * KDIM;
    const float* w3 = W + (size_t)(3 * 16 + m) * KDIM;

    v8f acc0 = {}, acc1 = {}, acc2 = {}, acc3 = {};

    // K = 0..63 : self features (x)
#pragma unroll
    for (int kk = 0; kk < 64; kk += 4) {
        const int k0 = kk + ko;
        v2f a = ld2(xrow + k0);
        acc0 = wmma_f32k4(a, ld2(w0 + k0), acc0);
        acc1 = wmma_f32k4(a, ld2(w1 + k0), acc1);
        acc2 = wmma_f32k4(a, ld2(w2 + k0), acc2);
        acc3 = wmma_f32k4(a, ld2(w3 + k0), acc3);
    }
    // K = 64..127 : mean-aggregated neighbor features
#pragma unroll
    for (int kk = 64; kk < 128; kk += 4) {
        const int k0 = kk + ko;
        v2f a = ld2(arow + (k0 - 64));
        acc0 = wmma_f32k4(a, ld2(w0 + k0), acc0);
        acc1 = wmma_f32k4(a, ld2(w1 + k0), acc1);
        acc2 = wmma_f32k4(a, ld2(w2 + k0), acc2);
        acc3 = wmma_f32k4(a, ld2(w3 + k0), acc3);
    }

    // Store: C/D layout — VGPR r: lanes 0-15 -> M=r, lanes 16-31 -> M=r+8
    const float b0 = bias[0 * 16 + m];
    const float b1 = bias[1 * 16 + m];
    const float b2 = bias[2 * 16 + m];
    const float b3 = bias[3 * 16 + m];
    const int mhi = (lane >> 4) << 3;
#pragma unroll
    for (int r = 0; r < 8; ++r) {
        const int row = tile * 16 + r + mhi;
        float* orow = out + (size_t)row * OUT_CH;
        orow[0 * 16 + m] = acc0[r] + b0;
        orow[1 * 16 + m] = acc1[r] + b1;
        orow[2 * 16 + m] = acc2[r] + b2;
        orow[3 * 16 + m] = acc3[r] + b3;
    }
}

// ---------------------------------------------------------------------------
// Scalar fallback for rows not covered by full 16-row tiles (not launched for
// N=50000 which is a multiple of 16, kept for generality).
// ---------------------------------------------------------------------------
__global__ void sage_gemm_rem(const float* __restrict__ x,
                              const float* __restrict__ aggr,
                              const float* __restrict__ W,
                              const float* __restrict__ bias,
                              float* __restrict__ out,
                              int start, int num_nodes) {
    int idx = blockIdx.x * blockDim.x + threadIdx.x;
    int row = start + (idx >> 6);
    int oc  = idx & 63;
    if (row >= num_nodes) return;
    float s = bias[oc];
    const float* w = W + (size_t)oc * KDIM;
    for (int k = 0; k < IN_CH; ++k) s += x[(size_t)row * IN_CH + k] * w[k];
    for (int k = 0; k < IN_CH; ++k) s += aggr[(size_t)row * IN_CH + k] * w[IN_CH + k];
    out[(size_t)row * OUT_CH + oc] = s;
}

extern "C" void kernel_launch(void* const* d_in, const int* in_sizes, int n_in,
                              void* d_out, int out_size, void* d_ws, size_t ws_size,
                              hipStream_t stream) {
    const float* x    = (const float*)d_in[0];
    const int*   edge = (const int*)d_in[1];
    const float* W    = (const float*)d_in[2];
    const float* bias = (const float*)d_in[3];
    float*       out  = (float*)d_out;

    const int num_nodes = in_sizes[0] / IN_CH;
    const int num_edges = in_sizes[1] / 2;

    float* aggr = (float*)d_ws;                       // [num_nodes * 64]
    float* deg  = aggr + (size_t)num_nodes * IN_CH;   // [num_nodes]

    // 1) zero aggr + deg (contiguous in ws)
    {
        long long total = (long long)num_nodes * IN_CH + num_nodes;
        int blocks = (int)((total + 255) / 256);
        sage_zero_ws<<<blocks, 256, 0, stream>>>((float*)d_ws, total);
    }
    // 2) edge scatter
    {
        long long threads = (long long)num_edges * 16;
        int blocks = (int)((threads + 255) / 256);
        sage_scatter<<<blocks, 256, 0, stream>>>(x, edge, aggr, deg, num_edges);
    }
    // 3) mean normalization
    {
        long long total = (long long)num_nodes * IN_CH;
        int blocks = (int)((total + 255) / 256);
        sage_normalize<<<blocks, 256, 0, stream>>>(aggr, deg, total);
    }
    // 4) WMMA GEMM: 8 waves / block, one 16x64 tile per wave
    {
        int tiles = num_nodes / 16;
        if (tiles > 0) {
            int wavesPerBlock = 8;
            int blocks = (tiles + wavesPerBlock - 1) / wavesPerBlock;
            sage_gemm_wmma<<<blocks, wavesPerBlock * 32, 0, stream>>>(
                x, aggr, W, bias, out, tiles);
        }
        int rem_start = tiles * 16;
        int rem = num_nodes - rem_start;
        if (rem > 0) {
            int threads = rem * OUT_CH;
            int blocks = (threads + 255) / 256;
            sage_gemm_rem<<<blocks, 256, 0, stream>>>(
                x, aggr, W, bias, out, rem_start, num_nodes);
        }
    }
}